// CIF_Forced_65343632441387
// MI455X (gfx1250) — compile-verified
//
#include <hip/hip_runtime.h>
#include <hip/hip_bf16.h>

#define BSZ 32
#define T   2048
#define D   768
#define L   8
#define KW  5
#define NROWS (BSZ * T)   // 65536

typedef __attribute__((ext_vector_type(2))) float v2f;
typedef __attribute__((ext_vector_type(8))) float v8f;

// ---------------------------------------------------------------------------
// Kernel 1: fold Linear(D,1) into the Conv1d weights.
//   v[k][di] = sum_do lin_w[do] * conv_w[do, di, k]      (conv_w: [D,D,5])
//   c0       = lin_b + sum_do lin_w[do] * conv_b[do]
// Blocks 0..14 fold (3840 outputs, reads coalesced over (di,k)); block 15 = c0.
// ---------------------------------------------------------------------------
__global__ void fold_weights_kernel(const float* __restrict__ conv_w,
                                    const float* __restrict__ conv_b,
                                    const float* __restrict__ lin_w,
                                    const float* __restrict__ lin_b,
                                    float* __restrict__ vfold,
                                    float* __restrict__ c0) {
    const int tid = threadIdx.x;
    if (blockIdx.x < 15) {
        const int j = blockIdx.x * 256 + tid;       // flat (di*5 + k), 0..3839
        float acc = 0.f;
        for (int o = 0; o < D; ++o)
            acc = fmaf(lin_w[o], conv_w[(size_t)o * (D * KW) + j], acc);
        vfold[(j % KW) * D + (j / KW)] = acc;       // store as v[k][di]
    } else {
        __shared__ float red[256];
        float acc = 0.f;
        for (int i = tid; i < D; i += 256)
            acc = fmaf(lin_w[i], conv_b[i], acc);
        red[tid] = acc;
        __syncthreads();
        for (int s = 128; s > 0; s >>= 1) {
            if (tid < s) red[tid] += red[tid + s];
            __syncthreads();
        }
        if (tid == 0) c0[0] = red[0] + lin_b[0];
    }
}

// ---------------------------------------------------------------------------
// Kernel 2: per-row projections P[k][row] = dot(v[k], x[row, :]).
// One wave32 per row; v staged in LDS. This is the 192 MB streaming pass:
// all x reads are float4 (global_load_b128), v reads are ds_load_b128.
// ---------------------------------------------------------------------------
__global__ void row_proj_kernel(const float* __restrict__ x,
                                const float* __restrict__ vfold,
                                float* __restrict__ P) {
    __shared__ float vs[KW * D];                     // 15 KB
    const int tid = threadIdx.x;
    {
        const float4* __restrict__ vsrc = (const float4*)vfold;
        float4* vdst = (float4*)vs;
        for (int i = tid; i < (KW * D) / 4; i += 256) vdst[i] = vsrc[i];
    }
    __syncthreads();

    const int  wave = tid >> 5;
    const int  lane = tid & 31;
    const long row  = (long)blockIdx.x * 8 + wave;   // 0..65535
    const float4* __restrict__ xr = (const float4*)(x + row * D);
    const float4* __restrict__ v4 = (const float4*)vs;

    float p0 = 0.f, p1 = 0.f, p2 = 0.f, p3 = 0.f, p4 = 0.f;
    #pragma unroll
    for (int it = 0; it < (D / 4) / 32; ++it) {      // 6 iterations
        const int f = it * 32 + lane;                // float4 index within row
        const float4 xv = xr[f];
        #define ACC(pk, plane)                                        \
            { const float4 vv = v4[(plane) * (D / 4) + f];            \
              pk = fmaf(vv.x, xv.x, pk); pk = fmaf(vv.y, xv.y, pk);   \
              pk = fmaf(vv.z, xv.z, pk); pk = fmaf(vv.w, xv.w, pk); }
        ACC(p0, 0) ACC(p1, 1) ACC(p2, 2) ACC(p3, 3) ACC(p4, 4)
        #undef ACC
    }
    #pragma unroll
    for (int off = 16; off > 0; off >>= 1) {
        p0 += __shfl_xor(p0, off);
        p1 += __shfl_xor(p1, off);
        p2 += __shfl_xor(p2, off);
        p3 += __shfl_xor(p3, off);
        p4 += __shfl_xor(p4, off);
    }
    if (lane == 0) {
        P[0 * NROWS + row] = p0;
        P[1 * NROWS + row] = p1;
        P[2 * NROWS + row] = p2;
        P[3 * NROWS + row] = p3;
        P[4 * NROWS + row] = p4;
    }
}

// ---------------------------------------------------------------------------
// Kernel 3: per batch — alphas (sigmoid of stencil of P, masked by len),
// block scan for cumsum + total, normalize to L, emit CIF overlap weights in
// PAIRED layout: Wt2[b][t/2][m][0..1] = (w[t][m], w[t+1][m]), rows m=8..15
// zero. This makes the WMMA A-fragment a single 64-bit load.
// ---------------------------------------------------------------------------
__global__ void cif_weights_kernel(const float* __restrict__ P,
                                   const float* __restrict__ c0p,
                                   const int*   __restrict__ lens,
                                   float* __restrict__ Wt2) {
    const int b   = blockIdx.x;
    const int tid = threadIdx.x;
    __shared__ float scan[256];

    const float c0  = c0p[0];
    const int   len = lens[b];
    const int   t0  = tid * 8;                       // contiguous 8-chunk per thread

    float a[8];
    float s = 0.f;
    #pragma unroll
    for (int j = 0; j < 8; ++j) {
        const int t = t0 + j;
        float z = c0;
        #pragma unroll
        for (int k = 0; k < KW; ++k) {
            const int u = t + k - 2;                 // SAME padding: OOB rows give 0
            if (u >= 0 && u < T)
                z += P[(size_t)k * NROWS + (size_t)b * T + u];
        }
        float al = 1.f / (1.f + __expf(-z));
        al = (t < len) ? al : 0.f;
        a[j] = al;
        s += al;
    }

    scan[tid] = s;
    __syncthreads();
    for (int off = 1; off < 256; off <<= 1) {        // Hillis-Steele inclusive scan
        const float v = (tid >= off) ? scan[tid - off] : 0.f;
        __syncthreads();
        scan[tid] += v;
        __syncthreads();
    }
    const float total  = scan[255];
    const float prefix = scan[tid] - s;              // exclusive prefix
    const float scale  = (float)L / total;

    float Aprev = prefix * scale;
    float* __restrict__ wbase = Wt2 + ((size_t)b * (T / 2) + (t0 >> 1)) * 32;
    for (int pr = 0; pr < 4; ++pr) {                 // 4 t-pairs per thread
        const float Amid = Aprev + a[2 * pr] * scale;
        const float Aend = Amid + a[2 * pr + 1] * scale;
        float we[16], wo[16];
        #pragma unroll
        for (int l = 0; l < L; ++l) {
            const float fl = (float)l, fh = (float)(l + 1);
            we[l] = fmaxf(fminf(Amid, fh) - fmaxf(Aprev, fl), 0.f);
            wo[l] = fmaxf(fminf(Aend, fh) - fmaxf(Amid,  fl), 0.f);
        }
        #pragma unroll
        for (int l = L; l < 16; ++l) { we[l] = 0.f; wo[l] = 0.f; }
        float4* wp = (float4*)(wbase + pr * 32);     // interleave (even,odd) per m
        #pragma unroll
        for (int q = 0; q < 8; ++q)
            wp[q] = make_float4(we[2 * q], wo[2 * q], we[2 * q + 1], wo[2 * q + 1]);
        Aprev = Aend;
    }
}

// ---------------------------------------------------------------------------
// Kernel 4: out[b, 0:8, :] = W(16 x T) @ x_b(T x D) via V_WMMA_F32_16X16X4_F32.
// 8 waves/block share one batch; the batch's weight matrix is staged in LDS
// in two 64 KB chunks (paired layout), so the A-fragment is one ds_load_b64.
// A frag layout: VGPR0 = K {k0, k0+2}, VGPR1 = K {k0+1, k0+3}, M = lane & 15.
// ---------------------------------------------------------------------------
__global__ void wmma_contract_kernel(const float* __restrict__ Wt2,
                                     const float* __restrict__ x,
                                     float* __restrict__ out) {
    __shared__ float aw[16384];                      // 64 KB: 1024 t (512 pairs) x 16 m
    const int tid  = threadIdx.x;
    const int lane = tid & 31;
    const int wave = tid >> 5;
    const int b    = blockIdx.x / 6;
    const int tg   = blockIdx.x % 6;
    const int n0   = (tg * 8 + wave) * 16;           // N-tile start in D
    const int m    = lane & 15;                      // M for A frag / N for B frag
    const int ks   = lane >> 4;                      // K-pair select (0 or 1)

    const float* __restrict__ xb = x + (size_t)b * T * D;

    v8f acc = {0.f, 0.f, 0.f, 0.f, 0.f, 0.f, 0.f, 0.f};

    for (int h = 0; h < 2; ++h) {                    // two 1024-t chunks
        __syncthreads();
        {
            const float4* __restrict__ src =
                (const float4*)(Wt2 + (size_t)b * T * 16 + (size_t)h * 16384);
            float4* dst = (float4*)aw;
            for (int i = tid; i < 4096; i += 256) dst[i] = src[i];
        }
        __syncthreads();

        const int kbase = h * 1024;
        #pragma unroll 4
        for (int kk = 0; kk < 1024; kk += 4) {
            const int p  = (kk >> 1) + ks;           // local pair index
            const int ka = kbase + kk + 2 * ks;      // global t of first K row
            const v2f av = *(const v2f*)&aw[p * 32 + 2 * m];   // ds_load_b64
            v2f bv;
            bv.x = xb[(size_t)ka * D + n0 + m];
            bv.y = xb[(size_t)(ka + 1) * D + n0 + m];
            acc = __builtin_amdgcn_wmma_f32_16x16x4_f32(
                      false, av, false, bv, (short)0, acc, false, false);
        }
    }

    if (lane < 16) {
        #pragma unroll
        for (int r = 0; r < 8; ++r)                  // rows 0..7 live in lanes 0..15
            out[((size_t)b * L + r) * D + n0 + m] = acc[r];
    }
}

// ---------------------------------------------------------------------------
// Host launcher
// ---------------------------------------------------------------------------
extern "C" void kernel_launch(void* const* d_in, const int* in_sizes, int n_in,
                              void* d_out, int out_size, void* d_ws, size_t ws_size,
                              hipStream_t stream) {
    const float* x      = (const float*)d_in[0];   // (32, 2048, 768)
    const int*   lens   = (const int*)  d_in[1];   // (32,)
    const float* conv_w = (const float*)d_in[2];   // (768, 768, 5)
    const float* conv_b = (const float*)d_in[3];   // (768,)
    const float* lin_w  = (const float*)d_in[4];   // (1, 768)
    const float* lin_b  = (const float*)d_in[5];   // (1,)
    float* out = (float*)d_out;                    // (32, 8, 768)

    float* ws    = (float*)d_ws;
    float* vfold = ws;                              // 3840 floats  v[k][di]
    float* c0    = ws + 3840;                       // 1 float
    float* P     = ws + 4096;                       // 5 * 65536 floats
    float* Wt2   = ws + 4096 + KW * NROWS;          // 32*2048*16 floats (paired)

    fold_weights_kernel <<<16,   256, 0, stream>>>(conv_w, conv_b, lin_w, lin_b, vfold, c0);
    row_proj_kernel     <<<8192, 256, 0, stream>>>(x, vfold, P);
    cif_weights_kernel  <<<BSZ,  256, 0, stream>>>(P, c0, lens, Wt2);
    wmma_contract_kernel<<<BSZ * 6, 256, 0, stream>>>(Wt2, x, out);
}